// HashEmbedder_36000415875355
// MI455X (gfx1250) — compile-verified
//
#include <hip/hip_runtime.h>
#include <stdint.h>

// Instant-NGP hash-grid encoder for gfx1250 (MI455X).
//
// Roofline: ~144 MB of HBM traffic (~6us at 23.3 TB/s) but ~1 GB of 8B gathers
// that must be served from L2 (table is 4 MB << 192 MB L2). Strategy:
//  - one thread per point, all 16 levels fully unrolled with compile-time
//    resolutions -> 8 global_load_b64 gathers in flight per level
//  - incremental hashing: 2 u32 muls per level instead of 24
//  - mask folded onto the 6 hash components (XOR commutes with AND)
//  - x streamed with TH_LOAD_NT, out written with TH_STORE_NT so the
//    write-once/read-once streams don't evict the hot embedding table from L2
//  - each lane writes exactly one 128 B output cacheline via 8x b128 NT stores

#define NGP_THREADS 256

typedef float v4f __attribute__((ext_vector_type(4)));  // native clang vector
                                                        // (nontemporal-builtin OK)

__global__ __launch_bounds__(NGP_THREADS)
void hashgrid_embed_gfx1250(const float* __restrict__ x,
                            const float* __restrict__ emb,
                            float* __restrict__ out,
                            int n) {
    const int i = blockIdx.x * NGP_THREADS + threadIdx.x;
    if (i >= n) return;

    // Stream the point array ahead of the wavefront (global_prefetch_b8).
    if (i + 8192 < n) {
        __builtin_prefetch(x + 3u * (size_t)(i + 8192), 0, 1);
    }

    // 12 contiguous bytes per point; non-temporal (read-once stream).
    const float* __restrict__ xp = x + 3 * (size_t)i;
    const float px = __builtin_nontemporal_load(xp + 0);
    const float py = __builtin_nontemporal_load(xp + 1);
    const float pz = __builtin_nontemporal_load(xp + 2);

    const float2* __restrict__ E = reinterpret_cast<const float2*>(emb);

    constexpr uint32_t P1 = 2654435761u;
    constexpr uint32_t P2 = 805459861u;
    constexpr uint32_t M  = (1u << 19) - 1u;   // HASHMAP_SIZE - 1

    // res_l = floor(16 * 2^(l/3)), b = exp(ln(512/16)/15) = cbrt(2).
    const float kRes[16] = { 16.f,  20.f,  25.f,  32.f,  40.f,  50.f,  64.f,  80.f,
                            101.f, 128.f, 161.f, 203.f, 256.f, 322.f, 406.f, 512.f };

    float acc[32];

#pragma unroll
    for (int l = 0; l < 16; ++l) {
        const float r = kRes[l];
        const float sx = px * r, sy = py * r, sz = pz * r;
        const float fx = floorf(sx), fy = floorf(sy), fz = floorf(sz);
        const float wx = sx - fx, wy = sy - fy, wz = sz - fz;

        const uint32_t cx = (uint32_t)fx;
        const uint32_t cy = (uint32_t)fy;
        const uint32_t cz = (uint32_t)fz;

        // Incremental hash components; mask each component (XOR commutes with
        // AND, adds don't -> mask after the adds). 6 ANDs instead of 8.
        const uint32_t hx0 = cx & M,              hx1 = (cx + 1u) & M;
        const uint32_t gy  = cy * P1;
        const uint32_t gz  = cz * P2;
        const uint32_t hy0 = gy & M,              hy1 = (gy + P1) & M;
        const uint32_t hz0 = gz & M,              hz1 = (gz + P2) & M;

        // Shared XY partials (4 XORs), then 8 final XORs.
        const uint32_t t00 = hx0 ^ hy0, t01 = hx0 ^ hy1;
        const uint32_t t10 = hx1 ^ hy0, t11 = hx1 ^ hy1;
        const uint32_t h000 = t00 ^ hz0, h001 = t00 ^ hz1;
        const uint32_t h010 = t01 ^ hz0, h011 = t01 ^ hz1;
        const uint32_t h100 = t10 ^ hz0, h101 = t10 ^ hz1;
        const uint32_t h110 = t11 ^ hz0, h111 = t11 ^ hz1;

        // Issue all 8 gathers before any consumer; all L2-resident.
        const float2 e000 = E[h000];
        const float2 e001 = E[h001];
        const float2 e010 = E[h010];
        const float2 e011 = E[h011];
        const float2 e100 = E[h100];
        const float2 e101 = E[h101];
        const float2 e110 = E[h110];
        const float2 e111 = E[h111];

        // Trilinear corner weights (dx outer, dy, dz inner enumeration).
        const float ux = 1.f - wx, uy = 1.f - wy, uz = 1.f - wz;
        const float a00 = ux * uy, a01 = ux * wy, a10 = wx * uy, a11 = wx * wy;
        const float c000 = a00 * uz, c001 = a00 * wz;
        const float c010 = a01 * uz, c011 = a01 * wz;
        const float c100 = a10 * uz, c101 = a10 * wz;
        const float c110 = a11 * uz, c111 = a11 * wz;

        // Two independent 4-deep FMA chains per feature (half the serial depth).
        float f0a = c000 * e000.x;
        float f0b = c100 * e100.x;
        float f1a = c000 * e000.y;
        float f1b = c100 * e100.y;
        f0a = fmaf(c001, e001.x, f0a);  f1a = fmaf(c001, e001.y, f1a);
        f0b = fmaf(c101, e101.x, f0b);  f1b = fmaf(c101, e101.y, f1b);
        f0a = fmaf(c010, e010.x, f0a);  f1a = fmaf(c010, e010.y, f1a);
        f0b = fmaf(c110, e110.x, f0b);  f1b = fmaf(c110, e110.y, f1b);
        f0a = fmaf(c011, e011.x, f0a);  f1a = fmaf(c011, e011.y, f1a);
        f0b = fmaf(c111, e111.x, f0b);  f1b = fmaf(c111, e111.y, f1b);

        acc[2 * l + 0] = f0a + f0b;
        acc[2 * l + 1] = f1a + f1b;
    }

    // One full 128 B cacheline per lane; non-temporal (write-once stream,
    // keep the embedding table resident in L2).
    v4f* __restrict__ o = reinterpret_cast<v4f*>(out + 32 * (size_t)i);
#pragma unroll
    for (int k = 0; k < 8; ++k) {
        v4f v;
        v.x = acc[4 * k + 0];
        v.y = acc[4 * k + 1];
        v.z = acc[4 * k + 2];
        v.w = acc[4 * k + 3];
        __builtin_nontemporal_store(v, o + k);
    }
}

extern "C" void kernel_launch(void* const* d_in, const int* in_sizes, int n_in,
                              void* d_out, int out_size, void* d_ws, size_t ws_size,
                              hipStream_t stream) {
    const float* x   = (const float*)d_in[0];   // [N,3] float32
    const float* emb = (const float*)d_in[1];   // [2^19, 2] float32
    float*       out = (float*)d_out;           // [N,32] float32

    const int n = in_sizes[0] / 3;
    const int blocks = (n + NGP_THREADS - 1) / NGP_THREADS;

    hipLaunchKernelGGL(hashgrid_embed_gfx1250,
                       dim3(blocks), dim3(NGP_THREADS), 0, stream,
                       x, emb, out, n);
}